// WeightedGCNLayer_21414706938338
// MI455X (gfx1250) — compile-verified
//
#include <hip/hip_runtime.h>

typedef __attribute__((ext_vector_type(2))) float v2f;
typedef __attribute__((ext_vector_type(8))) float v8f;

#define GCN_N_NODES 100000
#define GCN_N_EDGES 1600000
#define GCN_FEATS   32

// ---------------------------------------------------------------------------
// Kernel 1: zero the f32 accumulator in workspace (float4 stores, fully
// coalesced; 3.2M floats = 800K threads). Runs every call -> deterministic.
// ---------------------------------------------------------------------------
__global__ void gcn_zero_acc(float* __restrict__ acc, int n_float4) {
  int i = blockIdx.x * blockDim.x + threadIdx.x;
  if (i < n_float4) {
    ((float4*)acc)[i] = make_float4(0.f, 0.f, 0.f, 0.f);
  }
}

// ---------------------------------------------------------------------------
// Kernel 2: u_mul_e + scatter-sum.
// 8 lanes cooperate per edge; each lane handles 4 consecutive features via a
// float4 gather (one 128B line per edge across a wave-quarter) and 4 f32
// global atomics. Everything is L2-resident (h=12.8MB), so this is bound by
// L2 atomic throughput -- the minimum possible without sorting edges.
// ---------------------------------------------------------------------------
__global__ void gcn_edge_scatter(const float* __restrict__ h,
                                 const float* __restrict__ ew,
                                 const int*   __restrict__ src,
                                 const int*   __restrict__ dst,
                                 float*       __restrict__ acc) {
  long long idx = (long long)blockIdx.x * blockDim.x + threadIdx.x;
  if (idx >= (long long)GCN_N_EDGES * 8) return;
  int e = (int)(idx >> 3);          // edge index
  int c = (int)(idx & 7) << 2;      // feature chunk start (0,4,...,28)

  int   s = src[e];
  int   d = dst[e];
  float w = ew[e];

  float4 hv = *(const float4*)(h + (long long)s * GCN_FEATS + c);
  float* p = acc + (long long)d * GCN_FEATS + c;
  atomicAdd(p + 0, hv.x * w);
  atomicAdd(p + 1, hv.y * w);
  atomicAdd(p + 2, hv.z * w);
  atomicAdd(p + 3, hv.w * w);
}

// ---------------------------------------------------------------------------
// Kernel 3: out = acc @ W^T + bias via V_WMMA_F32_16X16X4_F32 (full f32).
// One wave computes a 16x32 output tile: two 16x16 accumulators, 8 K-steps
// of K=4 -> 16 WMMAs per wave. 6250 tiles total (100000 % 16 == 0).
//
// A (16x4, MxK) per-lane layout (ISA 7.12.2): lanes 0-15 hold K=0,1 in
// VGPR0/1, lanes 16-31 hold K=2,3  -> v2f load at k = kb + 2*(lane>>4).
// B (4x16, KxN): row-per-VGPR striped across lanes, mirrored K split.
// C/D (16x16): VGPR v <-> row v + 8*(lane>>4), col lane&15.
// Bias is pre-loaded into the C accumulator.
// ---------------------------------------------------------------------------
__global__ void gcn_gemm_wmma(const float* __restrict__ acc,
                              const float* __restrict__ W,
                              const float* __restrict__ bias,
                              float*       __restrict__ out) {
  const int lane = threadIdx.x & 31;
  const int wave = threadIdx.x >> 5;
  const int tile = blockIdx.x * 8 + wave;          // blockDim.x == 256 -> 8 waves
  if (tile >= GCN_N_NODES / 16) return;            // wave-uniform: EXEC stays all-1s

  const int m0   = tile * 16;
  const int half = lane >> 4;                      // 0: K=0,1 / rows 0-7 ; 1: K=2,3 / rows 8-15
  const int mn   = lane & 15;                      // A row / B-C-D column

  // Fold bias into C.
  v8f c0, c1;
  const float bv0 = bias[mn];
  const float bv1 = bias[16 + mn];
#pragma unroll
  for (int v = 0; v < 8; ++v) { c0[v] = bv0; c1[v] = bv1; }

  const float* arow = acc + (long long)(m0 + mn) * GCN_FEATS;
  const float* b0row = W + (long long)mn * GCN_FEATS;         // B[k][n] = W[n][k], N-tile 0
  const float* b1row = W + (long long)(16 + mn) * GCN_FEATS;  // N-tile 1

#pragma unroll
  for (int kb = 0; kb < GCN_FEATS; kb += 4) {
    const int k = kb + half * 2;
    v2f a  = *(const v2f*)(arow  + k);
    v2f b0 = *(const v2f*)(b0row + k);
    v2f b1 = *(const v2f*)(b1row + k);
    // 8 args: (neg_a, A, neg_b, B, c_mod, C, reuse_a, reuse_b)
    c0 = __builtin_amdgcn_wmma_f32_16x16x4_f32(false, a, false, b0, (short)0, c0, false, false);
    c1 = __builtin_amdgcn_wmma_f32_16x16x4_f32(false, a, false, b1, (short)0, c1, false, false);
  }

  // Store D: lane (half,mn) + VGPR v -> out[m0 + v + 8*half][{mn, 16+mn}]
#pragma unroll
  for (int v = 0; v < 8; ++v) {
    const int row = m0 + v + 8 * half;
    out[(long long)row * GCN_FEATS + mn]      = c0[v];
    out[(long long)row * GCN_FEATS + 16 + mn] = c1[v];
  }
}

// ---------------------------------------------------------------------------
extern "C" void kernel_launch(void* const* d_in, const int* in_sizes, int n_in,
                              void* d_out, int out_size, void* d_ws, size_t ws_size,
                              hipStream_t stream) {
  const float* h    = (const float*)d_in[0];   // [100000, 32]
  const float* ew   = (const float*)d_in[1];   // [1600000, 1]
  const int*   src  = (const int*)  d_in[2];   // [1600000]
  const int*   dst  = (const int*)  d_in[3];   // [1600000]
  const float* W    = (const float*)d_in[4];   // [32, 32]
  const float* bias = (const float*)d_in[5];   // [32]
  float*       out  = (float*)d_out;           // [100000, 32]
  float*       acc  = (float*)d_ws;            // scratch accumulator, 12.8 MB

  // 1) zero accumulator
  const int n_f4 = (GCN_N_NODES * GCN_FEATS) / 4;  // 800000
  gcn_zero_acc<<<(n_f4 + 255) / 256, 256, 0, stream>>>(acc, n_f4);

  // 2) gather * eweight -> scatter-add
  const long long ework = (long long)GCN_N_EDGES * 8;  // 12.8M threads
  gcn_edge_scatter<<<(int)((ework + 255) / 256), 256, 0, stream>>>(h, ew, src, dst, acc);

  // 3) WMMA GEMM + bias
  const int tiles = GCN_N_NODES / 16;  // 6250
  gcn_gemm_wmma<<<(tiles + 7) / 8, 256, 0, stream>>>(acc, W, bias, out);
}